// GATLayer_41412074668164
// MI455X (gfx1250) — compile-verified
//
#include <hip/hip_runtime.h>
#include <hip/hip_bf16.h>
#include <math.h>

typedef __attribute__((ext_vector_type(2))) float v2f;
typedef __attribute__((ext_vector_type(8))) float v8f;

#define NEG_BIG (-9.0e15f)
#define BATCH 16
#define NNODE 1024
#define FDIM  64

__device__ __forceinline__ float lrelu(float v) { return v > 0.f ? v : 0.2f * v; }

// ---------------------------------------------------------------------------
// Kernel 1: h = x @ W.  One wave (32 threads) computes a 16x64 tile of h
// using v_wmma_f32_16x16x4_f32 (full fp32 precision matrix pipe).
// A (16x4 f32): lanes 0-15 hold M=lane, K={0,1}; lanes 16-31 hold K={2,3}.
// B (4x16 f32): lanes 0-15 hold N=lane, K={0,1}; lanes 16-31 hold K={2,3}.
// C (16x16 f32): VGPR v -> M=v (lanes 0-15) / M=v+8 (lanes 16-31), N=lane&15.
// ---------------------------------------------------------------------------
__global__ __launch_bounds__(32)
void gat_xw_kernel(const float* __restrict__ x, const float* __restrict__ W,
                   float* __restrict__ h) {
  const int tile = blockIdx.x;             // B*N/16 = 1024 tiles
  const int b    = tile >> 6;
  const int i0   = (tile & 63) << 4;
  const int lane = threadIdx.x;
  const int mrow = lane & 15;
  const int half = lane >> 4;

  v8f acc0 = {}, acc1 = {}, acc2 = {}, acc3 = {};
  const float* xrow = x + ((size_t)(b * NNODE + i0 + mrow)) * FDIM;

  for (int kc = 0; kc < 16; ++kc) {
    const int k = kc * 4 + half * 2;
    v2f A;
    A.x = xrow[k];
    A.y = xrow[k + 1];
    const float* w0 = W + (size_t)k * FDIM;
    const float* w1 = w0 + FDIM;
    v2f B0, B1, B2, B3;
    B0.x = w0[mrow];      B0.y = w1[mrow];
    B1.x = w0[16 + mrow]; B1.y = w1[16 + mrow];
    B2.x = w0[32 + mrow]; B2.y = w1[32 + mrow];
    B3.x = w0[48 + mrow]; B3.y = w1[48 + mrow];
    acc0 = __builtin_amdgcn_wmma_f32_16x16x4_f32(false, A, false, B0, (short)0, acc0, false, false);
    acc1 = __builtin_amdgcn_wmma_f32_16x16x4_f32(false, A, false, B1, (short)0, acc1, false, false);
    acc2 = __builtin_amdgcn_wmma_f32_16x16x4_f32(false, A, false, B2, (short)0, acc2, false, false);
    acc3 = __builtin_amdgcn_wmma_f32_16x16x4_f32(false, A, false, B3, (short)0, acc3, false, false);
  }

  float* hout = h + ((size_t)(b * NNODE + i0)) * FDIM;
  for (int v = 0; v < 8; ++v) {
    const int M = v + 8 * half;
    float* dst = hout + (size_t)M * FDIM;
    dst[mrow]      = acc0[v];
    dst[16 + mrow] = acc1[v];
    dst[32 + mrow] = acc2[v];
    dst[48 + mrow] = acc3[v];
  }
}

// ---------------------------------------------------------------------------
// Kernel 2: per-node scores  s_src = h . a[:64],  s_dst = h . a[64:]
// ---------------------------------------------------------------------------
__global__ __launch_bounds__(256)
void gat_scores_kernel(const float* __restrict__ h, const float* __restrict__ a,
                       float* __restrict__ s_src, float* __restrict__ s_dst) {
  const int row = blockIdx.x * 256 + threadIdx.x;   // 0..16383
  const float* hr = h + (size_t)row * FDIM;
  float s1 = 0.f, s2 = 0.f;
#pragma unroll 4
  for (int f = 0; f < FDIM; ++f) {
    const float v = hr[f];
    s1 += v * a[f];
    s2 += v * a[FDIM + f];
  }
  s_src[row] = s1;
  s_dst[row] = s2;
}

// ---------------------------------------------------------------------------
// Kernel 3: masked-softmax attention + h' = attn @ h + ELU.
// One 256-thread block (8 wave32) per (batch, 16-row tile).
//   Pass 1: online softmax stats (m, denom) per row, width-16 shuffle reduce.
//   Pass 2: waves split j into chunks of 4; attention built in registers,
//           aggregated with v_wmma_f32_16x16x4_f32; cross-wave LDS reduce.
// ---------------------------------------------------------------------------
__global__ __launch_bounds__(256)
void gat_attn_kernel(const int* __restrict__ adj, const float* __restrict__ h,
                     const float* __restrict__ s_src, const float* __restrict__ s_dst,
                     float* __restrict__ out) {
  __shared__ float lds_acc[8 * 16 * FDIM];   // 32 KB: per-wave partial tiles
  __shared__ float ssrc_sh[16];
  __shared__ float m_sh[16];
  __shared__ float inv_sh[16];

  const int tile = blockIdx.x;               // 1024 tiles
  const int b    = tile >> 6;
  const int i0   = (tile & 63) << 4;
  const int t    = threadIdx.x;
  const size_t rowbase = (size_t)(b * NNODE + i0);
  const float* sdb = s_dst + (size_t)b * NNODE;

  // ---- Pass 1: softmax statistics ----
  {
    const int r = t >> 4;                    // row within tile
    const int c = t & 15;                    // j stripe
    const float ssrc_i = s_src[rowbase + r];
    const int* arow = adj + (rowbase + r) * (size_t)NNODE;
    float m = -INFINITY, s = 0.f;
    for (int it = 0; it < NNODE / 16; ++it) {
      const int j = c + (it << 4);
      const float e = (arow[j] > 0) ? lrelu(ssrc_i + sdb[j]) : NEG_BIG;
      if (e > m) { s = s * expf(m - e) + 1.f; m = e; }
      else       { s += expf(e - m); }
    }
    // combine the 16 (m,s) pairs of this row (stays inside one wave32 half)
    for (int off = 8; off >= 1; off >>= 1) {
      const float m2 = __shfl_xor(m, off, 16);
      const float s2 = __shfl_xor(s, off, 16);
      const float nm = fmaxf(m, m2);
      s = s * expf(m - nm) + s2 * expf(m2 - nm);
      m = nm;
    }
    if (c == 0) { m_sh[r] = m; inv_sh[r] = 1.f / s; }
    if (t < 16) ssrc_sh[t] = s_src[rowbase + t];
  }
  __syncthreads();

  // ---- Pass 2: attn @ h via fp32 WMMA ----
  const int lane = t & 31;
  const int w    = t >> 5;
  const int mrow = lane & 15;
  const int half = lane >> 4;
  const float ssrc_i = ssrc_sh[mrow];
  const float mi     = m_sh[mrow];
  const float inv    = inv_sh[mrow];
  const int* arow  = adj + (rowbase + mrow) * (size_t)NNODE;
  const float* hb  = h + (size_t)b * NNODE * FDIM;

  v8f acc0 = {}, acc1 = {}, acc2 = {}, acc3 = {};
  for (int chunk = w; chunk < NNODE / 4; chunk += 8) {
    const int jb = chunk * 4 + half * 2;
    const int2   ad  = *(const int2*)(arow + jb);
    const float2 sd2 = *(const float2*)(sdb + jb);
    const float e0 = (ad.x > 0) ? lrelu(ssrc_i + sd2.x) : NEG_BIG;
    const float e1 = (ad.y > 0) ? lrelu(ssrc_i + sd2.y) : NEG_BIG;
    v2f A;
    A.x = expf(e0 - mi) * inv;
    A.y = expf(e1 - mi) * inv;
    const float* h0 = hb + (size_t)jb * FDIM;
    const float* h1 = h0 + FDIM;
    v2f B0, B1, B2, B3;
    B0.x = h0[mrow];      B0.y = h1[mrow];
    B1.x = h0[16 + mrow]; B1.y = h1[16 + mrow];
    B2.x = h0[32 + mrow]; B2.y = h1[32 + mrow];
    B3.x = h0[48 + mrow]; B3.y = h1[48 + mrow];
    acc0 = __builtin_amdgcn_wmma_f32_16x16x4_f32(false, A, false, B0, (short)0, acc0, false, false);
    acc1 = __builtin_amdgcn_wmma_f32_16x16x4_f32(false, A, false, B1, (short)0, acc1, false, false);
    acc2 = __builtin_amdgcn_wmma_f32_16x16x4_f32(false, A, false, B2, (short)0, acc2, false, false);
    acc3 = __builtin_amdgcn_wmma_f32_16x16x4_f32(false, A, false, B3, (short)0, acc3, false, false);
  }

  // park per-wave partial 16x64 tiles in LDS
  for (int v = 0; v < 8; ++v) {
    const int M = v + 8 * half;
    float* dst = &lds_acc[(w * 16 + M) * FDIM];
    dst[mrow]      = acc0[v];
    dst[16 + mrow] = acc1[v];
    dst[32 + mrow] = acc2[v];
    dst[48 + mrow] = acc3[v];
  }
  __syncthreads();

  // cross-wave reduce + ELU + store
  for (int e = t; e < 16 * FDIM; e += 256) {
    const int M = e >> 6, n = e & 63;
    float sum = 0.f;
#pragma unroll
    for (int ww = 0; ww < 8; ++ww) sum += lds_acc[(ww * 16 + M) * FDIM + n];
    out[(rowbase + M) * FDIM + n] = sum > 0.f ? sum : expm1f(sum);
  }
}

// ---------------------------------------------------------------------------
extern "C" void kernel_launch(void* const* d_in, const int* in_sizes, int n_in,
                              void* d_out, int out_size, void* d_ws, size_t ws_size,
                              hipStream_t stream) {
  const float* x   = (const float*)d_in[0];   // (16,1024,64) f32
  const int*   adj = (const int*)  d_in[1];   // (16,1024,1024) i32
  const float* W   = (const float*)d_in[2];   // (64,64) f32
  const float* a   = (const float*)d_in[3];   // (128,) f32
  float* out = (float*)d_out;                 // (16,1024,64) f32

  float* h     = (float*)d_ws;                            // 4 MB
  float* s_src = h + (size_t)BATCH * NNODE * FDIM;        // 64 KB
  float* s_dst = s_src + (size_t)BATCH * NNODE;           // 64 KB

  gat_xw_kernel<<<BATCH * NNODE / 16, 32, 0, stream>>>(x, W, h);
  gat_scores_kernel<<<BATCH * NNODE / 256, 256, 0, stream>>>(h, a, s_src, s_dst);
  gat_attn_kernel<<<BATCH * NNODE / 16, 256, 0, stream>>>(adj, h, s_src, s_dst, out);
}